// Euler_75299366633583
// MI455X (gfx1250) — compile-verified
//
#include <hip/hip_runtime.h>
#include <math.h>

typedef float v2f __attribute__((ext_vector_type(2)));
typedef float v4f __attribute__((ext_vector_type(4)));
typedef float v8f __attribute__((ext_vector_type(8)));

#define B_    32
#define C_    128
#define T_    64
#define V_    25
#define TV    1600            // T*V, contiguous per (b,c)
#define CTV   (C_ * TV)       // 204800
#define SEG   (B_ * CTV)      // 6553600 elements per segment
#define NSTEP 8

// ---------------------------------------------------------------------------
// Segment 0: z_shift[0] = z, z_cls[0] = z  (pure bandwidth, float4)
// ---------------------------------------------------------------------------
__global__ __launch_bounds__(256)
void Euler_seg0_copy(const float* __restrict__ z, float* __restrict__ out) {
    long i = (long)blockIdx.x * blockDim.x + threadIdx.x;   // float4 index
    v4f v = ((const v4f*)z)[i];
    ((v4f*)out)[i] = v;                      // z_shift segment 0
    ((v4f*)(out + 9L * SEG))[i] = v;         // z_cls   segment 0
}

// ---------------------------------------------------------------------------
// One Euler step: next = cur + tanh(W @ cur + b), fused stores of
// z_cls[step] and rolled+masked z_shift[step].
// Block = 256 threads = 8 waves. Block handles one (batch, 16-col) tile:
//   - cooperatively stage cur[b, 0:128, n0:n0+16] (8 KB) in LDS
//   - wave w computes output rows d0 = 16*w .. 16*w+15 via 32x WMMA f32 16x16x4
// ---------------------------------------------------------------------------
__global__ __launch_bounds__(256)
void Euler_step_kernel(const float* __restrict__ cur,   // [B,C,T,V] (prev z_cls seg)
                       const float* __restrict__ W,     // [C,C] row-major W[d][c]
                       const float* __restrict__ bias,  // [C]
                       float* __restrict__ out_cls,     // z_cls segment base
                       float* __restrict__ out_shift,   // z_shift segment base
                       int step) {
    __shared__ float ldsB[C_ * 16];          // [k][n] tile of cur, 8 KB

    const int tid = threadIdx.x;
    const int n0  = blockIdx.x * 16;         // column tile within TV
    const int b   = blockIdx.y;
    const float* curb = cur + (long)b * CTV;

    // ---- stage cur tile into LDS: each thread moves 8 contiguous floats ----
    {
        int k    = tid >> 1;                 // 0..127
        int noff = (tid & 1) * 8;            // 0 or 8
        const float* src = curb + (long)k * TV + n0 + noff;
        v4f a0 = *(const v4f*)(src);
        v4f a1 = *(const v4f*)(src + 4);
        *(v4f*)(&ldsB[k * 16 + noff])     = a0;
        *(v4f*)(&ldsB[k * 16 + noff + 4]) = a1;
    }
    __syncthreads();

    const int lane  = tid & 31;
    const int wave  = tid >> 5;
    const int d0    = wave * 16;             // output row tile
    const int laneN = lane & 15;             // N index for B/C/D frags, M for A
    const int kOff  = (lane >> 4) << 1;      // A/B K sub-offset: 0 (lo half) / 2 (hi half)

    // ---- K loop: D = W[d0:,k] x curTile[k,:] accumulated over K=128 ----
    v8f acc = {};
    const float* Wrow = W + (long)(d0 + laneN) * C_ + kOff;
    #pragma unroll 8
    for (int k4 = 0; k4 < C_; k4 += 4) {
        v2f a = *(const v2f*)(Wrow + k4);                     // A frag: W[d,k..k+1]
        v2f bf;
        bf.x = ldsB[(k4 + kOff)     * 16 + laneN];            // B frag: cur[k,n]
        bf.y = ldsB[(k4 + kOff + 1) * 16 + laneN];
        acc = __builtin_amdgcn_wmma_f32_16x16x4_f32(
                  /*neg_a=*/false, a, /*neg_b=*/false, bf,
                  /*c_mod=*/(short)0, acc, /*reuse_a=*/false, /*reuse_b=*/false);
    }

    // ---- fused epilogue: bias + tanh + residual, dual store ----
    // C/D layout: VGPR j -> M = j (lanes 0-15) / j+8 (lanes 16-31), N = lane%16
    const int  mAdd = (lane >> 4) << 3;      // 0 or 8
    const long obase = (long)b * CTV;
    const int  n = n0 + laneN;
    #pragma unroll
    for (int j = 0; j < 8; ++j) {
        int d = d0 + j + mAdd;
        float curv = ldsB[d * 16 + laneN];               // residual from LDS tile
        float val  = curv + tanhf(acc[j] + bias[d]);
        long rowo  = obase + (long)d * TV;
        out_cls[rowo + n] = val;
        // roll by +step frames along T with first `step` frames zeroed.
        // src frame t -> dst frame t+step if in range; the overflow frames
        // (t >= T-step) instead zero-fill dst frames [0, step) bijectively.
        int nd = n + step * V_;
        if (nd < TV) out_shift[rowo + nd] = val;
        else         out_shift[rowo + nd - TV] = 0.0f;
    }
}

// ---------------------------------------------------------------------------
extern "C" void kernel_launch(void* const* d_in, const int* in_sizes, int n_in,
                              void* d_out, int out_size, void* d_ws, size_t ws_size,
                              hipStream_t stream) {
    const float* z    = (const float*)d_in[0];   // (32,128,64,25)
    const float* W    = (const float*)d_in[1];   // (128,128)
    const float* bias = (const float*)d_in[2];   // (128,)
    float* out       = (float*)d_out;
    float* shiftBase = out;                      // z_shift: segments 0..8
    float* clsBase   = out + 9L * SEG;           // z_cls:   segments 0..8

    // segment 0 of both outputs = z
    Euler_seg0_copy<<<SEG / 4 / 256, 256, 0, stream>>>(z, out);

    // 8 sequential Euler steps; z_cls segments double as the recurrence buffer
    dim3 grid(TV / 16, B_);                      // 100 x 32 blocks
    for (int i = 1; i <= NSTEP; ++i) {
        const float* cur = clsBase + (long)(i - 1) * SEG;
        Euler_step_kernel<<<grid, 256, 0, stream>>>(
            cur, W, bias,
            clsBase + (long)i * SEG,
            shiftBase + (long)i * SEG,
            i);
    }
}